// DiffSAGEWrapper_41051297415238
// MI455X (gfx1250) — compile-verified
//
#include <hip/hip_runtime.h>
#include <hip/hip_bf16.h>

typedef __attribute__((ext_vector_type(2))) float v2f;
typedef __attribute__((ext_vector_type(8))) float v8f;

#define D 128

// ---------------- kernel 1: zero workspace ----------------
__global__ void zero_ws_kernel(float* __restrict__ p, int n) {
    int i = blockIdx.x * blockDim.x + threadIdx.x;
    if (i < n) p[i] = 0.0f;
}

// ---------------- kernel 2: edge scatter ----------------
// One wave32 per edge. Accumulate sum_src[dst] += x[src]; cnt[dst] += 1.
// (msg sum decomposes as sum(x[src]) - cnt*x[dst], applied in kernel 3.)
__global__ void edge_scatter_kernel(const long long* __restrict__ ei,
                                    const float* __restrict__ x,
                                    float* __restrict__ sum_src,
                                    float* __restrict__ cnt, int E) {
    int gid  = blockIdx.x * blockDim.x + threadIdx.x;
    int e    = gid >> 5;
    int lane = gid & 31;
    if (e >= E) return;
    int src = (int)ei[e];
    int dst = (int)ei[(size_t)E + e];
    float4 v = ((const float4*)(x + (size_t)src * D))[lane];
    float* dp = sum_src + (size_t)dst * D + lane * 4;
    atomicAdd(dp + 0, v.x);
    atomicAdd(dp + 1, v.y);
    atomicAdd(dp + 2, v.z);
    atomicAdd(dp + 3, v.w);
    if (lane == 0) atomicAdd(cnt + dst, 1.0f);
}

// ---------------- kernel 3: agg reconstruction + dual GEMM via f32 WMMA ----------------
// Block = 256 threads = 8 waves. Each block: 16 nodes. Each wave: one 16x16 output tile.
// out[m, n] = sum_k agg[m,k]*Wl[n,k] + sum_k x[m,k]*Wr[n,k] + bl[n]
__global__ void __launch_bounds__(256)
sage_wmma_kernel(const float* __restrict__ x,
                 const float* __restrict__ S,
                 const float* __restrict__ cnt,
                 const float* __restrict__ Wl,
                 const float* __restrict__ blv,
                 const float* __restrict__ Wr,
                 float* __restrict__ out, int N) {
    __shared__ float lx[16][D + 2];   // row pad: stride 130 floats breaks 64-bank alignment
    __shared__ float lagg[16][D + 2];

    const int node0 = blockIdx.x * 16;
    const int tid   = threadIdx.x;

    // Stage x tile and agg tile (16 x 128) into LDS: 512 float4 slots, 2 per thread.
    #pragma unroll
    for (int it = 0; it < 2; ++it) {
        int idx  = tid + it * 256;      // 0..511
        int row  = idx >> 5;            // 0..15
        int c4   = idx & 31;            // float4 column
        int node = node0 + row;
        float4 xv = make_float4(0.f, 0.f, 0.f, 0.f);
        float4 sv = make_float4(0.f, 0.f, 0.f, 0.f);
        float  c  = 0.f;
        if (node < N) {
            xv = ((const float4*)(x + (size_t)node * D))[c4];
            sv = ((const float4*)(S + (size_t)node * D))[c4];
            c  = cnt[node];
        }
        float rinv = 1.0f / fmaxf(c, 1.0f);
        int col = c4 * 4;
        lx[row][col + 0] = xv.x;  lx[row][col + 1] = xv.y;
        lx[row][col + 2] = xv.z;  lx[row][col + 3] = xv.w;
        lagg[row][col + 0] = (sv.x - c * xv.x) * rinv;
        lagg[row][col + 1] = (sv.y - c * xv.y) * rinv;
        lagg[row][col + 2] = (sv.z - c * xv.z) * rinv;
        lagg[row][col + 3] = (sv.w - c * xv.w) * rinv;
    }
    __syncthreads();

    const int wave = tid >> 5;
    const int lane = tid & 31;
    const int half = lane >> 4;     // 0: K pair {0,1}; 1: K pair {2,3}
    const int l16  = lane & 15;
    const int col0 = wave * 16;     // this wave's 16 output channels

    v8f acc = {0.f, 0.f, 0.f, 0.f, 0.f, 0.f, 0.f, 0.f};

    // B[k][n] = W[col0+n][k]; per ISA layout lane holds n=l16, k = 2*half + vgpr.
    const float* wlrow = Wl + (size_t)(col0 + l16) * D;
    const float* wrrow = Wr + (size_t)(col0 + l16) * D;

    #pragma unroll 4
    for (int k0 = 0; k0 < D; k0 += 4) {
        const int k = k0 + 2 * half;
        // A 16x4 fp32 layout: lanes 0-15 hold M=l16, K={k0,k0+1}; lanes 16-31 K={k0+2,k0+3}
        v2f a  = { lagg[l16][k], lagg[l16][k + 1] };
        v2f b  = { wlrow[k],     wlrow[k + 1]     };
        acc = __builtin_amdgcn_wmma_f32_16x16x4_f32(false, a, false, b,
                                                    (short)0, acc, false, false);
        v2f a2 = { lx[l16][k],   lx[l16][k + 1]   };
        v2f b2 = { wrrow[k],     wrrow[k + 1]     };
        acc = __builtin_amdgcn_wmma_f32_16x16x4_f32(false, a2, false, b2,
                                                    (short)0, acc, false, false);
    }

    // C/D 16x16 f32 layout: VGPR r -> M = r + 8*half, N = l16.
    const float bias = blv[col0 + l16];
    #pragma unroll
    for (int r = 0; r < 8; ++r) {
        int node = node0 + r + 8 * half;
        if (node < N) out[(size_t)node * D + col0 + l16] = acc[r] + bias;
    }
}

extern "C" void kernel_launch(void* const* d_in, const int* in_sizes, int n_in,
                              void* d_out, int out_size, void* d_ws, size_t ws_size,
                              hipStream_t stream) {
    const float*     x   = (const float*)d_in[0];
    const long long* ei  = (const long long*)d_in[1];  // int64 per reference
    const float*     Wl  = (const float*)d_in[2];
    const float*     bl  = (const float*)d_in[3];
    const float*     Wr  = (const float*)d_in[4];
    float*           out = (float*)d_out;

    const int N = in_sizes[0] / D;
    const int E = in_sizes[1] / 2;

    float* sum_src = (float*)d_ws;                       // N*128 floats
    float* cnt     = sum_src + (size_t)N * D;            // N floats

    // 1) zero accumulators (sum_src and cnt are contiguous)
    int ztot = N * (D + 1);
    zero_ws_kernel<<<(ztot + 255) / 256, 256, 0, stream>>>(sum_src, ztot);

    // 2) edge scatter: one wave32 per edge
    long long ethreads = (long long)E * 32;
    int eblocks = (int)((ethreads + 255) / 256);
    edge_scatter_kernel<<<eblocks, 256, 0, stream>>>(ei, x, sum_src, cnt, E);

    // 3) agg + dual GEMM with fp32 WMMA, one block per 16-node tile
    int gblocks = (N + 15) / 16;
    sage_wmma_kernel<<<gblocks, 256, 0, stream>>>(x, sum_src, cnt, Wl, bl, Wr, out, N);
}